// STD_30915174596800
// MI455X (gfx1250) — compile-verified
//
#include <hip/hip_runtime.h>
#include <cmath>

// FISTA solver for the reference problem, fp32 WMMA (V_WMMA_F32_16X16X4_F32).
// L=128, D=512, M=2048. Output = [A@x (L*M) | A (L*D)].

typedef __attribute__((ext_vector_type(2))) float v2f;
typedef __attribute__((ext_vector_type(8))) float v8f;

#define L_ 128
#define D_ 512
#define M_ 2048
#define TAIL 127
#define MAX_IT 200

static constexpr float LAM1 = 0.1f;
static constexpr float LAM2 = 0.1f;
static constexpr float EPSC = 1e-5f;

__device__ __forceinline__ v8f wmma4(v2f a, v2f b, v8f c) {
  // D = A(16x4,f32) * B(4x16,f32) + C(16x16,f32)
  return __builtin_amdgcn_wmma_f32_16x16x4_f32(false, a, false, b, (short)0, c,
                                               false, false);
}

__device__ __forceinline__ v8f vzero8() {
  v8f z = {0.f, 0.f, 0.f, 0.f, 0.f, 0.f, 0.f, 0.f};
  return z;
}

// Delta matrix entry (row k, col i): I - 2*sub1 + sub2, with [0,0]=0, [1,0]=-1
__device__ __forceinline__ float delta_elem(int k, int i) {
  float v = 0.f;
  if (k == i) v = 1.f;
  else if (k == i + 1) v = -2.f;
  else if (k == i + 2) v = 1.f;
  if (i == 0) {
    if (k == 0) v = 0.f;
    else if (k == 1) v = -1.f;
  }
  return v;
}

// ---------------- setup kernels ----------------

__global__ void k_init_small(float* __restrict__ DtDs, const float* __restrict__ y0,
                             float* __restrict__ ytail, int* __restrict__ done,
                             float* __restrict__ sc) {
  int t = threadIdx.x;  // 256 threads, 1 block
  for (int idx = t; idx < L_ * L_; idx += 256) {
    int i = idx >> 7, j = idx & (L_ - 1);
    int klo = (i > j) ? i : j;
    int khi = ((i < j) ? i : j) + 2;
    if (khi > L_ - 1) khi = L_ - 1;
    float s = 0.f;
    for (int k = klo; k <= khi; ++k) s += delta_elem(k, i) * delta_elem(k, j);
    DtDs[idx] = LAM1 * s;  // pre-scale by lambda_1
  }
  if (t < TAIL) ytail[t] = y0[M_ + t];  // y0[2048 .. 2174]
  if (t == TAIL) ytail[t] = 0.f;
  if (t == 0) { *done = 0; sc[0] = 0.f; sc[1] = 0.f; }
}

__global__ void k_init_big(const float* __restrict__ mask, const float* __restrict__ py,
                           float* __restrict__ AX, float* __restrict__ R,
                           float* __restrict__ A0, float* __restrict__ A1,
                           float* __restrict__ Bm) {
  int idx = blockIdx.x * 256 + threadIdx.x;  // grid = 1024*256 = L_*M_
  AX[idx] = 0.f;
  R[idx] = -mask[idx] * py[idx];  // mask*(0*x - part_y)
  if (idx < L_ * D_) { A0[idx] = 0.f; A1[idx] = 0.f; Bm[idx] = 0.f; }
}

// xT (M x D) = transpose of x (D x M): makes A@x B-fragments contiguous b64 loads.
__global__ void __launch_bounds__(256) k_transpose(const float* __restrict__ x,
                                                   float* __restrict__ xT) {
  __shared__ float tile[32][33];
  int bx = blockIdx.x & ((M_ / 32) - 1);  // m-tile (64 of them)
  int by = blockIdx.x / (M_ / 32);        // d-tile (16 of them)
  int tx = threadIdx.x & 31, ty = threadIdx.x >> 5;  // 32 x 8
#pragma unroll
  for (int r = 0; r < 32; r += 8)
    tile[ty + r][tx] = x[(by * 32 + ty + r) * M_ + bx * 32 + tx];
  __syncthreads();
#pragma unroll
  for (int r = 0; r < 32; r += 8)
    xT[(bx * 32 + ty + r) * D_ + by * 32 + tx] = tile[tx][ty + r];
}

// S = x @ x^T  (512x512, K=2048). One wave -> 16x64 tile. grid 32 blocks x 8 waves.
__global__ void __launch_bounds__(256) k_xxt(const float* __restrict__ x,
                                             float* __restrict__ S) {
  int lane = threadIdx.x & 31, w = threadIdx.x >> 5;
  int row0 = blockIdx.x * 16;  // 32 row tiles
  int col0 = w * 64;           // 8 waves * 64 = 512 cols
  int m = lane & 15, kb = (lane >> 4) << 1;
  v8f acc[4] = {vzero8(), vzero8(), vzero8(), vzero8()};
  const float* ap = x + (row0 + m) * M_;
  for (int k0 = 0; k0 < M_; k0 += 4) {
    v2f a = *(const v2f*)(ap + k0 + kb);
#pragma unroll
    for (int t = 0; t < 4; ++t) {
      v2f b = *(const v2f*)(x + (col0 + t * 16 + m) * M_ + k0 + kb);  // B = x^T
      acc[t] = wmma4(a, b, acc[t]);
    }
  }
  int hi = lane >> 4, n = lane & 15;
#pragma unroll
  for (int t = 0; t < 4; ++t)
#pragma unroll
    for (int r = 0; r < 8; ++r)
      S[(row0 + r + 8 * hi) * D_ + col0 + t * 16 + n] = acc[t][r];
}

// Power iteration: eig_x = maxeig(S), eig_s = maxeig(DtDs) = lambda1*maxeig(DtD).
// tau = 1/(2*(eig_x + eig_s)); thr = tau*lam2.
__global__ void k_power(const float* __restrict__ S, const float* __restrict__ DtDs,
                        float* __restrict__ sc) {
  __shared__ float sv[D_];
  __shared__ float red[D_];
  int t = threadIdx.x;  // 512 threads
  sv[t] = 1.0f;
  __syncthreads();
  float eig_x = 0.f;
  for (int it = 0; it < 100; ++it) {
    float wv = 0.f;
    const float* row = S + t * D_;
    for (int j = 0; j < D_; ++j) wv += row[j] * sv[j];
    red[t] = wv * wv;
    __syncthreads();
    for (int s = 256; s > 0; s >>= 1) {
      if (t < s) red[t] += red[t + s];
      __syncthreads();
    }
    float nrm = sqrtf(red[0]) + 1e-30f;
    __syncthreads();
    sv[t] = wv / nrm;
    eig_x = nrm;
    __syncthreads();
  }
  // DtDs (128x128): init with alternating signs (top mode of 2nd-difference op)
  if (t < L_) sv[t] = (t & 1) ? -1.f : 1.f;
  __syncthreads();
  float eig_s = 0.f;
  for (int it = 0; it < 100; ++it) {
    float wv = 0.f;
    if (t < L_) {
      const float* row = DtDs + t * L_;
      for (int j = 0; j < L_; ++j) wv += row[j] * sv[j];
    }
    red[t] = (t < L_) ? wv * wv : 0.f;
    __syncthreads();
    for (int s = 256; s > 0; s >>= 1) {
      if (t < s) red[t] += red[t + s];
      __syncthreads();
    }
    float nrm = sqrtf(red[0]) + 1e-30f;
    __syncthreads();
    if (t < L_) sv[t] = wv / nrm;
    eig_s = nrm;
    __syncthreads();
  }
  if (t == 0) {
    float tau = 1.0f / (2.0f * (eig_x + eig_s));
    float lam2 = (tau * LAM2 > 0.1f) ? (0.1f / tau) : LAM2;
    sc[0] = tau;
    sc[1] = tau * lam2;
  }
}

// ---------------- per-iteration kernels ----------------

// G = DtDs@A0 + R@x^T ; A1 = soft(B - tau*G); B = A0 + c*(A1 - A0)
// grid: 8 blocks (row tiles) x 8 waves (64-col slabs) -> 128x512
__global__ void __launch_bounds__(256) k_grad(
    const float* __restrict__ R, const float* __restrict__ x,
    const float* __restrict__ DtDs, const float* __restrict__ A0,
    float* __restrict__ A1, float* __restrict__ Bm,
    const float* __restrict__ sc, const int* __restrict__ done, float cmom) {
  if (*done) return;  // uniform -> EXEC stays all-1s for WMMA
  int lane = threadIdx.x & 31, w = threadIdx.x >> 5;
  int row0 = blockIdx.x * 16;
  int col0 = w * 64;
  int m = lane & 15, kb = (lane >> 4) << 1;
  v8f acc[4] = {vzero8(), vzero8(), vzero8(), vzero8()};
  // R @ x^T  (K = M_): B[k][n] = x[col+n][k] -> contiguous b64 per fragment
  const float* ap = R + (row0 + m) * M_;
  for (int k0 = 0; k0 < M_; k0 += 4) {
    v2f a = *(const v2f*)(ap + k0 + kb);
#pragma unroll
    for (int t = 0; t < 4; ++t) {
      v2f b = *(const v2f*)(x + (col0 + t * 16 + m) * M_ + k0 + kb);
      acc[t] = wmma4(a, b, acc[t]);
    }
  }
  // DtDs @ A0  (K = L_, small)
  const float* dp = DtDs + (row0 + m) * L_;
  for (int k0 = 0; k0 < L_; k0 += 4) {
    v2f a = *(const v2f*)(dp + k0 + kb);
#pragma unroll
    for (int t = 0; t < 4; ++t) {
      const float* bp = A0 + (k0 + kb) * D_ + col0 + t * 16 + m;
      v2f b;
      b.x = bp[0];
      b.y = bp[D_];
      acc[t] = wmma4(a, b, acc[t]);
    }
  }
  float tau = sc[0], thr = sc[1];
  int hi = lane >> 4, n = lane & 15;
#pragma unroll
  for (int t = 0; t < 4; ++t) {
#pragma unroll
    for (int r = 0; r < 8; ++r) {
      int idx = (row0 + r + 8 * hi) * D_ + col0 + t * 16 + n;
      float g = acc[t][r];
      float bo = Bm[idx];
      float ao = A0[idx];
      float z = bo - tau * g;
      float an = copysignf(fmaxf(fabsf(z) - thr, 0.f), z);
      A1[idx] = an;
      Bm[idx] = ao + cmom * (an - ao);
    }
  }
}

__global__ void k_copy(float* __restrict__ A0, const float* __restrict__ A1,
                       const int* __restrict__ done) {
  if (*done) return;
  int idx = blockIdx.x * 256 + threadIdx.x;  // grid = 256*256 = L_*D_
  A0[idx] = A1[idx];
}

// AX = A0 @ x ; R = mask*(AX - part_y). Uses xT so B-fragments are b64 loads.
// grid: 32 blocks (64-col slabs) x 8 waves (row tiles) -> 128x2048
__global__ void __launch_bounds__(256) k_ax(
    const float* __restrict__ A0, const float* __restrict__ xT,
    const float* __restrict__ mask, const float* __restrict__ py,
    float* __restrict__ AX, float* __restrict__ R, const int* __restrict__ done) {
  if (*done) return;
  int lane = threadIdx.x & 31, w = threadIdx.x >> 5;
  int col0 = blockIdx.x * 64;
  int row0 = w * 16;
  int m = lane & 15, kb = (lane >> 4) << 1;
  v8f acc[4] = {vzero8(), vzero8(), vzero8(), vzero8()};
  const float* ap = A0 + (row0 + m) * D_;
  for (int k0 = 0; k0 < D_; k0 += 4) {
    v2f a = *(const v2f*)(ap + k0 + kb);
#pragma unroll
    for (int t = 0; t < 4; ++t) {
      // B[k][n] = x[k][col+n] = xT[col+n][k] -> contiguous pair
      v2f b = *(const v2f*)(xT + (col0 + t * 16 + m) * D_ + k0 + kb);
      acc[t] = wmma4(a, b, acc[t]);
    }
  }
  int hi = lane >> 4, n = lane & 15;
#pragma unroll
  for (int t = 0; t < 4; ++t) {
#pragma unroll
    for (int r = 0; r < 8; ++r) {
      int idx = (row0 + r + 8 * hi) * M_ + col0 + t * 16 + n;
      float v = acc[t][r];
      AX[idx] = v;
      R[idx] = mask[idx] * (v - py[idx]);
    }
  }
}

// Tail of hankel_to_scalar(AX) (entries 2048..2174), nrmse vs previous tail,
// set done flag, update stored tail.
__global__ void k_hankel(const float* __restrict__ AX, float* __restrict__ ytail,
                         int* __restrict__ done) {
  if (*done) return;
  __shared__ float s_num[128];
  __shared__ float s_den[128];
  int t = threadIdx.x;  // 128 threads, 1 block
  float axv = 0.f;
  if (t < TAIL) {
    int nn = M_ + t;  // 2048+t
    float s = 0.f;
    for (int i = t + 1; i < L_; ++i) s += AX[i * M_ + (nn - i)];
    axv = s / (float)(TAIL - t);  // count = min(n+1, 2175-n, 128)
  }
  float yv = (t < TAIL) ? ytail[t] : 0.f;
  float d = yv - axv;
  s_num[t] = (t < TAIL) ? d * d : 0.f;
  s_den[t] = (t < TAIL) ? yv * yv : 0.f;
  __syncthreads();
  for (int s = 64; s > 0; s >>= 1) {
    if (t < s) { s_num[t] += s_num[t + s]; s_den[t] += s_den[t + s]; }
    __syncthreads();
  }
  if (t < TAIL) ytail[t] = axv;
  if (t == 0) {
    float conv = sqrtf(s_num[0]) / (sqrtf(s_den[0]) + 1e-30f);
    if (conv <= EPSC) *done = 1;
  }
}

__global__ void k_out(const float* __restrict__ AX, const float* __restrict__ A0,
                      float* __restrict__ out) {
  int idx = blockIdx.x * 256 + threadIdx.x;  // grid = 1280*256 = L_*M_ + L_*D_
  if (idx < L_ * M_) out[idx] = AX[idx];
  else out[idx] = A0[idx - L_ * M_];
}

// ---------------- host ----------------

extern "C" void kernel_launch(void* const* d_in, const int* in_sizes, int n_in,
                              void* d_out, int out_size, void* d_ws, size_t ws_size,
                              hipStream_t stream) {
  const float* x    = (const float*)d_in[0];  // (D, M)
  const float* py   = (const float*)d_in[1];  // (L, M)
  const float* mask = (const float*)d_in[2];  // (L, M)
  const float* y0   = (const float*)d_in[3];  // (L+M-1,)
  (void)in_sizes; (void)n_in; (void)out_size; (void)ws_size;

  float* ws    = (float*)d_ws;
  float* DtDs  = ws;                 // 16384   (= lambda1 * Delta^T Delta)
  float* AX    = DtDs + L_ * L_;     // 262144  (also aliased as S=x@x^T in setup)
  float* R     = AX + L_ * M_;       // 262144
  float* A0    = R + L_ * M_;        // 65536
  float* A1    = A0 + L_ * D_;       // 65536
  float* Bm    = A1 + L_ * D_;       // 65536
  float* xT    = Bm + L_ * D_;       // 1048576 (M x D transpose of x)
  float* ytail = xT + M_ * D_;       // 128
  float* sc    = ytail + 128;        // [tau, thr]
  int*   done  = (int*)(sc + 8);

  // setup: DtDs, y-tail, done; xT; S = x@x^T (in AX slot); power iteration ->
  // tau/thr; then zero A/B/AX and seed R = -mask*part_y (AX slot reused).
  k_init_small<<<1, 256, 0, stream>>>(DtDs, y0, ytail, done, sc);
  k_transpose<<<(D_ / 32) * (M_ / 32), 256, 0, stream>>>(x, xT);
  k_xxt<<<32, 256, 0, stream>>>(x, AX);
  k_power<<<1, 512, 0, stream>>>(AX, DtDs, sc);
  k_init_big<<<(L_ * M_) / 256, 256, 0, stream>>>(mask, py, AX, R, A0, A1, Bm);

  float a = 1.f;
  for (int it = 0; it < MAX_IT; ++it) {
    float an = 0.5f * (1.f + sqrtf(1.f + 4.f * a * a));
    float c = (a - 1.f) / an;  // momentum coefficient (deterministic sequence)
    a = an;
    k_grad<<<8, 256, 0, stream>>>(R, x, DtDs, A0, A1, Bm, sc, done, c);
    k_copy<<<(L_ * D_) / 256, 256, 0, stream>>>(A0, A1, done);
    k_ax<<<32, 256, 0, stream>>>(A0, xT, mask, py, AX, R, done);
    k_hankel<<<1, 128, 0, stream>>>(AX, ytail, done);
  }

  k_out<<<(L_ * M_ + L_ * D_) / 256, 256, 0, stream>>>(AX, A0, (float*)d_out);
}